// RGAT_2989297238409
// MI455X (gfx1250) — compile-verified
//
#include <hip/hip_runtime.h>

#define N_ENT   100000
#define N_REL   50
#define LATDIM  64
#define N_EDGES 1000000

typedef __attribute__((ext_vector_type(2))) float v2f;
typedef __attribute__((ext_vector_type(8))) float v8f;

// ---- monotone float<->uint encoding so integer atomicMax orders floats ----
__device__ __forceinline__ unsigned enc_f(float f) {
    unsigned s = __float_as_uint(f);
    return (s & 0x80000000u) ? ~s : (s | 0x80000000u);
}
__device__ __forceinline__ float dec_f(unsigned u) {
    return (u & 0x80000000u) ? __uint_as_float(u ^ 0x80000000u) : __uint_as_float(~u);
}

// ============================================================================
// Wr[r, k] = sum_j W[k, j] * rel[r, j]   (r<50, k<128, j<64)
// i.e. Wr = rel[50x64] @ W^T[64x128], computed with V_WMMA_F32_16X16X4_F32.
// One wave per 16x16 output tile; 16 K-steps of 4.
// ============================================================================
__global__ void rgat_wr_wmma(const float* __restrict__ rel,   // [N_REL, 64]
                             const float* __restrict__ W,     // [128, 64] row-major
                             float* __restrict__ Wr)          // [N_REL, 128]
{
    const int lane = threadIdx.x;        // 0..31 (wave32)
    const int lo = lane & 15;
    const int hi = lane >> 4;            // 0 or 1
    const int tileM = blockIdx.x >> 3;   // 0..3  (rows of Wr, 16 each, 50 padded to 64)
    const int tileN = blockIdx.x & 7;    // 0..7  (cols of Wr, 16 each)
    const int r0 = tileM * 16;
    const int n0 = tileN * 16;

    const int  rA = r0 + lo;
    const bool rvalid = (rA < N_REL);
    const int  rclamp = rvalid ? rA : 0;       // branchless pad -> zero A rows

    v8f c = {};
    #pragma unroll
    for (int t = 0; t < 16; ++t) {
        const int kb = 4 * t;
        // A (16x4 f32): lanes0-15 M=0..15; VGPR0: K=0(lo half)/K=2(hi half); VGPR1: K=1/K=3
        v2f a;
        float a0 = rel[(size_t)rclamp * LATDIM + kb + 2 * hi + 0];
        float a1 = rel[(size_t)rclamp * LATDIM + kb + 2 * hi + 1];
        a[0] = rvalid ? a0 : 0.0f;
        a[1] = rvalid ? a1 : 0.0f;
        // B (4x16 f32): B[j, n] = W[n*64 + j]; VGPR v holds K=v (lanes0-15), K=v+2 (lanes16-31)
        v2f b;
        b[0] = W[(size_t)(n0 + lo) * LATDIM + kb + 2 * hi + 0];
        b[1] = W[(size_t)(n0 + lo) * LATDIM + kb + 2 * hi + 1];
        c = __builtin_amdgcn_wmma_f32_16x16x4_f32(false, a, false, b, (short)0, c,
                                                  false, false);
    }
    // D (16x16 f32): VGPR v -> row v (lanes0-15) / v+8 (lanes16-31), col = lo
    #pragma unroll
    for (int vg = 0; vg < 8; ++vg) {
        int m = vg + 8 * hi;
        if (r0 + m < N_REL) Wr[(size_t)(r0 + m) * 128 + n0 + lo] = c[vg];
    }
}

// ============================================================================
// Zero a contiguous f32/u32 region (segmax | segsum | agg laid out contiguously)
// ============================================================================
__global__ void rgat_zero(float* __restrict__ p, long long n)
{
    long long i = (long long)blockIdx.x * blockDim.x + threadIdx.x;
    long long s = (long long)gridDim.x * blockDim.x;
    for (; i < n; i += s) p[i] = 0.0f;
}

// ============================================================================
// Per-edge logit: e = leaky( ent[h]·Wr[r,0:64] + ent[t]·Wr[r,64:128] )
// wave-per-edge, float2 per lane, shfl reduce; segment max via encoded atomicMax
// ============================================================================
__global__ void rgat_logits(const float* __restrict__ ent,
                            const float* __restrict__ Wr,
                            const int* __restrict__ head,
                            const int* __restrict__ tail,
                            const int* __restrict__ etype,
                            float* __restrict__ ebuf,
                            unsigned* __restrict__ segmax)
{
    const int lane = threadIdx.x & 31;
    const int wid  = (blockIdx.x * blockDim.x + threadIdx.x) >> 5;
    const int nw   = (gridDim.x * blockDim.x) >> 5;
    for (int e = wid; e < N_EDGES; e += nw) {
        const int h = head[e], t = tail[e], r = etype[e];
        const float2 eh = *(const float2*)(ent + (size_t)h * LATDIM + lane * 2);
        const float2 tt = *(const float2*)(ent + (size_t)t * LATDIM + lane * 2);
        const float2 wh = *(const float2*)(Wr + (size_t)r * 128 + lane * 2);
        const float2 wt = *(const float2*)(Wr + (size_t)r * 128 + 64 + lane * 2);
        float p = eh.x * wh.x + eh.y * wh.y + tt.x * wt.x + tt.y * wt.y;
        #pragma unroll
        for (int m = 16; m >= 1; m >>= 1) p += __shfl_xor(p, m, 32);
        if (lane == 0) {
            float l = p > 0.0f ? p : 0.2f * p;    // leaky_relu(0.2)
            ebuf[e] = l;
            atomicMax(&segmax[h], enc_f(l));
        }
    }
}

// ============================================================================
// w = exp(e - segmax[head]); segsum[head] += w   (thread-per-edge)
// ============================================================================
__global__ void rgat_expsum(const int* __restrict__ head,
                            const unsigned* __restrict__ segmax,
                            float* __restrict__ ebuf,
                            float* __restrict__ segsum)
{
    int i = blockIdx.x * blockDim.x + threadIdx.x;
    int s = gridDim.x * blockDim.x;
    for (int e = i; e < N_EDGES; e += s) {
        int h = head[e];
        float w = __expf(ebuf[e] - dec_f(segmax[h]));
        ebuf[e] = w;
        atomicAdd(&segsum[h], w);
    }
}

// ============================================================================
// agg[head] += attn * ent[tail]   (wave-per-edge, float2 per lane, f32 atomics;
// both agg and ent are L2-resident on MI455X's 192MB L2)
// ============================================================================
__global__ void rgat_agg(const float* __restrict__ ent,
                         const int* __restrict__ head,
                         const int* __restrict__ tail,
                         const float* __restrict__ ebuf,
                         const float* __restrict__ segsum,
                         float* __restrict__ agg)
{
    const int lane = threadIdx.x & 31;
    const int wid  = (blockIdx.x * blockDim.x + threadIdx.x) >> 5;
    const int nw   = (gridDim.x * blockDim.x) >> 5;
    for (int e = wid; e < N_EDGES; e += nw) {
        const int h = head[e], t = tail[e];
        const float attn = ebuf[e] / segsum[h];
        const float2 v = *(const float2*)(ent + (size_t)t * LATDIM + lane * 2);
        float* dst = agg + (size_t)h * LATDIM + lane * 2;
        atomicAdd(dst + 0, v.x * attn);
        atomicAdd(dst + 1, v.y * attn);
    }
}

// ============================================================================
// v = agg + entIn; n = v / max(||v||, eps)
// hop0: entOut = n.   hop1(final): out = 0.25*E0 + 0.5*n1 + n   (wave-per-entity)
// ============================================================================
__global__ void rgat_norm(const float* __restrict__ agg,
                          const float* __restrict__ entIn,
                          float* __restrict__ entOut,
                          const float* __restrict__ E0,
                          const float* __restrict__ n1,
                          float* __restrict__ outp,
                          int finalHop)
{
    const int lane = threadIdx.x & 31;
    const int wid  = (blockIdx.x * blockDim.x + threadIdx.x) >> 5;
    const int nw   = (gridDim.x * blockDim.x) >> 5;
    for (int i = wid; i < N_ENT; i += nw) {
        const size_t base = (size_t)i * LATDIM + lane * 2;
        const float2 a = *(const float2*)(agg + base);
        const float2 x = *(const float2*)(entIn + base);
        float2 v = make_float2(a.x + x.x, a.y + x.y);
        float p = v.x * v.x + v.y * v.y;
        #pragma unroll
        for (int m = 16; m >= 1; m >>= 1) p += __shfl_xor(p, m, 32);
        const float scale = 1.0f / fmaxf(sqrtf(p), 1e-12f);
        const float2 n = make_float2(v.x * scale, v.y * scale);
        if (!finalHop) {
            *(float2*)(entOut + base) = n;
        } else {
            const float2 e0 = *(const float2*)(E0 + base);
            const float2 m1 = *(const float2*)(n1 + base);
            float2 o = make_float2(0.25f * e0.x + 0.5f * m1.x + n.x,
                                   0.25f * e0.y + 0.5f * m1.y + n.y);
            *(float2*)(outp + base) = o;
        }
    }
}

// ============================================================================
extern "C" void kernel_launch(void* const* d_in, const int* in_sizes, int n_in,
                              void* d_out, int out_size, void* d_ws, size_t ws_size,
                              hipStream_t stream)
{
    const float* ent0 = (const float*)d_in[0];   // [N_ENT, 64]
    const float* rel  = (const float*)d_in[1];   // [N_REL, 64]
    const float* W    = (const float*)d_in[2];   // [128, 64]
    const int* head   = (const int*)d_in[3];     // [E]
    const int* tail   = (const int*)d_in[4];     // [E]
    const int* etype  = (const int*)d_in[5];     // [E]
    float* out = (float*)d_out;

    // Workspace layout (bytes). segmax|segsum|agg are contiguous for one zero-fill.
    char* ws = (char*)d_ws;
    float*    Wr     = (float*)(ws + 0);           //  6400 f32
    float*    ebuf   = (float*)(ws + 32768);       //  1e6  f32
    unsigned* segmax = (unsigned*)(ws + 4032768);  //  1e5  u32 (enc; 0 == identity)
    float*    segsum = (float*)(ws + 4432768);     //  1e5  f32
    float*    aggv   = (float*)(ws + 4832768);     //  6.4e6 f32
    float*    ent1   = (float*)(ws + 30432768);    //  6.4e6 f32  (total ~56 MB)

    // Wr depends only on W and rel -> compute once (WMMA f32 16x16x4)
    rgat_wr_wmma<<<32, 32, 0, stream>>>(rel, W, Wr);

    const int WB = 2048;  // 2048 blocks * 8 waves = 16384 waves for edge loops
    for (int hop = 0; hop < 2; ++hop) {
        const float* entIn = (hop == 0) ? ent0 : ent1;
        rgat_zero  <<<2048, 256, 0, stream>>>((float*)segmax,
                                              (long long)N_ENT * (2 + LATDIM));
        rgat_logits<<<WB,   256, 0, stream>>>(entIn, Wr, head, tail, etype, ebuf, segmax);
        rgat_expsum<<<2048, 256, 0, stream>>>(head, segmax, ebuf, segsum);
        rgat_agg   <<<WB,   256, 0, stream>>>(entIn, head, tail, ebuf, segsum, aggv);
        rgat_norm  <<<1024, 256, 0, stream>>>(aggv, entIn, ent1, ent0, ent1, out, hop);
    }
}